// LabelAttention_13348758356158
// MI455X (gfx1250) — compile-verified
//
#include <hip/hip_runtime.h>

typedef __attribute__((ext_vector_type(2))) float v2f;
typedef __attribute__((ext_vector_type(8))) float v8f;

namespace {
constexpr int kL   = 48;
constexpr int kDK  = 512;
constexpr int kInQ = 1024;
constexpr int kInK = 1024;
constexpr int kVD  = 1024;
constexpr int kB   = 8;
constexpr int kS   = 512;
constexpr int kRows = kB * kS;            // 4096 (b,s) rows

// workspace layout (in floats)
constexpr size_t WS_KPROJ  = 0;                       // 48*512
constexpr size_t WS_W      = WS_KPROJ + (size_t)kL * kDK;  // 1024
constexpr size_t WS_C      = WS_W + kInQ;             // 1
constexpr size_t WS_SCORES = 25664;                   // 4096, 16-float aligned
} // namespace

// ---------------------------------------------------------------------------
// Stage 1: kproj[l,d] = sum_i K[l,i] * WK[l,i,d] + bK[l,d]
// one block per label; streams WK (100.7 MB) coalesced across d.
// ---------------------------------------------------------------------------
__global__ void kproj_kernel(const float* __restrict__ K,
                             const float* __restrict__ WK,
                             const float* __restrict__ bK,
                             float* __restrict__ kproj) {
  __shared__ float kl[kInK];
  const int l = blockIdx.x;
  const int t = threadIdx.x;            // blockDim = 512
  kl[t]       = K[l * kInK + t];
  kl[t + 512] = K[l * kInK + t + 512];
  __syncthreads();
  const float* wk = WK + (size_t)l * kInK * kDK + t;
  float acc = bK[l * kDK + t];
#pragma unroll 8
  for (int i = 0; i < kInK; ++i)
    acc = fmaf(kl[i], wk[(size_t)i * kDK], acc);
  kproj[l * kDK + t] = acc;
}

// ---------------------------------------------------------------------------
// Stage 2: w[i] = sum_{l,d} WQ[l,i,d] * kproj[l,d]
// one block per i; streams WQ (100.7 MB); kproj (96 KB) stays in L2.
// Deterministic fixed-order tree reduction (no float atomics).
// ---------------------------------------------------------------------------
__global__ void wvec_kernel(const float* __restrict__ WQ,
                            const float* __restrict__ kproj,
                            float* __restrict__ w) {
  __shared__ float red[256];
  const int i = blockIdx.x;
  const int t = threadIdx.x;            // blockDim = 256, 2 d's per thread
  float acc = 0.f;
  for (int l = 0; l < kL; ++l) {
    const float2* row = (const float2*)(WQ + ((size_t)l * kInQ + i) * kDK);
    const float2* kp  = (const float2*)(kproj + (size_t)l * kDK);
    const float2 rv = row[t];
    const float2 kv = kp[t];
    acc = fmaf(rv.x, kv.x, acc);
    acc = fmaf(rv.y, kv.y, acc);
  }
  red[t] = acc;
  __syncthreads();
  for (int off = 128; off > 0; off >>= 1) {
    if (t < off) red[t] += red[t + off];
    __syncthreads();
  }
  if (t == 0) w[i] = red[0];
}

// ---------------------------------------------------------------------------
// Stage 3: c = sum_{l,d} bQ[l,d] * kproj[l,d]   (tiny, single block)
// ---------------------------------------------------------------------------
__global__ void cscal_kernel(const float* __restrict__ bQ,
                             const float* __restrict__ kproj,
                             float* __restrict__ c) {
  __shared__ float red[kDK];
  const int t = threadIdx.x;            // blockDim = 512
  float acc = 0.f;
  for (int l = 0; l < kL; ++l)
    acc = fmaf(bQ[l * kDK + t], kproj[l * kDK + t], acc);
  red[t] = acc;
  __syncthreads();
  for (int off = kDK / 2; off > 0; off >>= 1) {
    if (t < off) red[t] += red[t + off];
    __syncthreads();
  }
  if (t == 0) c[0] = red[0];
}

// ---------------------------------------------------------------------------
// Stage 4: scores[b,s] = (Q[b,s,:]·w + c)/sqrt(512) + 48*(-1e9)*mask[b,s]
// via V_WMMA_F32_16X16X4_F32: A = 16 Q rows, B = w broadcast over all 16
// N-columns (so every column of D is the same dot product; no extra traffic).
// A layout (ISA 7.12.2): lane -> M = lane&15; VGPR0/1 hold K = {0,1} for
// lanes 0-15 and K = {2,3} for lanes 16-31. B assumed with the symmetric
// striping (K = vgpr + 2*(lane>=16), N = lane&15). Per lane both operands
// need the consecutive pair {k_lo, k_lo+1} -> one b64 load each.
// D extracted per documented C/D layout: lane 0 holds M=0..7, lane 16 M=8..15.
// ---------------------------------------------------------------------------
__global__ void scores_wmma_kernel(const float* __restrict__ Q,
                                   const float* __restrict__ w,
                                   const float* __restrict__ cptr,
                                   const int* __restrict__ mask,
                                   float* __restrict__ scores) {
  __shared__ float wl[kInQ];
  const int t = threadIdx.x;            // blockDim = 256 -> 8 waves
  for (int j = t; j < kInQ; j += 256) wl[j] = w[j];
  __syncthreads();

  const int lane  = t & 31;
  const int wave  = t >> 5;
  const int m0    = (blockIdx.x * 8 + wave) * 16;   // first of 16 rows
  const int row   = m0 + (lane & 15);               // A-matrix M = lane&15
  const int khalf = (lane < 16) ? 0 : 2;            // K offset per lane half

  const float* qrow = Q + (size_t)row * kInQ + khalf;
  const float* wrow = wl + khalf;

  v8f acc = {0.f, 0.f, 0.f, 0.f, 0.f, 0.f, 0.f, 0.f};
#pragma unroll 4
  for (int k0 = 0; k0 < kInQ; k0 += 4) {
    v2f a, bb;
    a.x  = qrow[k0];     a.y  = qrow[k0 + 1];       // b64 global load
    bb.x = wrow[k0];     bb.y = wrow[k0 + 1];       // b64 LDS read
    acc = __builtin_amdgcn_wmma_f32_16x16x4_f32(
        /*neg_a=*/false, a, /*neg_b=*/false, bb,
        /*c_mod=*/(short)0, acc, /*reuse_a=*/false, /*reuse_b=*/false);
  }

  const float cv    = cptr[0];
  const float scale = 0.044194173824159216f;        // 1/sqrt(512)
  if ((lane & 15) == 0) {                           // lanes 0 and 16
    const int rbase = m0 + ((lane == 16) ? 8 : 0);
#pragma unroll
    for (int x = 0; x < 8; ++x) {
      const int r = rbase + x;
      scores[r] = (acc[x] + cv) * scale + (-4.8e10f) * (float)mask[r];
    }
  }
}

// ---------------------------------------------------------------------------
// Stage 5: per batch: softmax over S, then context[b,:] = attn · V[b]
// writes both outputs: out[0:8192) = context, out[8192:12288) = attn
// ---------------------------------------------------------------------------
__global__ void softmax_context_kernel(const float* __restrict__ scores,
                                       const float* __restrict__ V,
                                       float* __restrict__ out) {
  __shared__ float red[kS];
  __shared__ float attn[kS];
  const int b = blockIdx.x;
  const int t = threadIdx.x;            // blockDim = 512
  const float sc = scores[b * kS + t];

  red[t] = sc;
  __syncthreads();
  for (int off = kS / 2; off > 0; off >>= 1) {
    if (t < off) red[t] = fmaxf(red[t], red[t + off]);
    __syncthreads();
  }
  const float m = red[0];
  __syncthreads();

  const float e = __expf(sc - m);
  red[t] = e;
  __syncthreads();
  for (int off = kS / 2; off > 0; off >>= 1) {
    if (t < off) red[t] += red[t + off];
    __syncthreads();
  }
  const float a = e / red[0];
  attn[t] = a;
  out[kB * kVD + b * kS + t] = a;       // attn output
  __syncthreads();

  const float* vb = V + (size_t)b * kS * kVD;
  float acc0 = 0.f, acc1 = 0.f;
  for (int s = 0; s < kS; ++s) {
    const float av = attn[s];
    acc0 = fmaf(av, vb[(size_t)s * kVD + t],       acc0);
    acc1 = fmaf(av, vb[(size_t)s * kVD + t + 512], acc1);
  }
  out[b * kVD + t]       = acc0;        // context output
  out[b * kVD + t + 512] = acc1;
}

// ---------------------------------------------------------------------------
extern "C" void kernel_launch(void* const* d_in, const int* in_sizes, int n_in,
                              void* d_out, int out_size, void* d_ws, size_t ws_size,
                              hipStream_t stream) {
  (void)in_sizes; (void)n_in; (void)out_size; (void)ws_size;
  const float* Q    = (const float*)d_in[0];
  const float* K    = (const float*)d_in[1];
  const float* V    = (const float*)d_in[2];
  const int*   mask = (const int*)  d_in[3];
  const float* WQ   = (const float*)d_in[4];
  const float* bQ   = (const float*)d_in[5];
  const float* WK   = (const float*)d_in[6];
  const float* bK   = (const float*)d_in[7];

  float* ws     = (float*)d_ws;
  float* kproj  = ws + WS_KPROJ;
  float* w      = ws + WS_W;
  float* c      = ws + WS_C;
  float* scores = ws + WS_SCORES;
  float* out    = (float*)d_out;

  kproj_kernel<<<dim3(kL), dim3(kDK), 0, stream>>>(K, WK, bK, kproj);
  wvec_kernel<<<dim3(kInQ), dim3(256), 0, stream>>>(WQ, kproj, w);
  cscal_kernel<<<dim3(1), dim3(kDK), 0, stream>>>(bQ, kproj, c);
  scores_wmma_kernel<<<dim3(kRows / 128), dim3(256), 0, stream>>>(Q, w, c, mask, scores);
  softmax_context_kernel<<<dim3(kB), dim3(kS), 0, stream>>>(scores, V, out);
}